// DynamicNeuralTuringMachine_4080218931367
// MI455X (gfx1250) — compile-verified
//
#include <hip/hip_runtime.h>
#include <math.h>

// dims (match reference)
#define B_ 64
#define T_ 32
#define F_ 128
#define H_ 256
#define N_ 512
#define C_ 96
#define A_ 32
#define M_ 128   // C_+A_
#define O_ 10
#define EPSF 1e-8f

typedef float v2f __attribute__((ext_vector_type(2)));
typedef float v8f __attribute__((ext_vector_type(8)));

// ---- fp32 WMMA 16x16x4 fragment helpers (CDNA5 VGPR layouts, wave32) ----
// A 16x4 (MxK): lane<16 -> M=lane, VGPR0=K0,VGPR1=K1 ; lane>=16 -> K2,K3
__device__ __forceinline__ v2f ld_a(const float* W, int ld, int m0, int k0, int lane) {
  int m = m0 + (lane & 15);
  int k = k0 + ((lane >> 4) << 1);
  v2f a; a.x = W[(size_t)m * ld + k]; a.y = W[(size_t)m * ld + k + 1]; return a;
}
// B 4x16 (KxN): col = lane&15, VGPR v holds K = v + 2*(lane>>4)
__device__ __forceinline__ v2f ld_b(const float* S, int ld, int k0, int n0, int lane) {
  int n = n0 + (lane & 15);
  int k = k0 + ((lane >> 4) << 1);
  v2f b; b.x = S[(size_t)k * ld + n]; b.y = S[(size_t)(k + 1) * ld + n]; return b;
}
__device__ __forceinline__ v8f wmma4(v2f a, v2f b, v8f c) {
  // (neg_a, A, neg_b, B, c_mod, C, reuse_a, reuse_b)
  return __builtin_amdgcn_wmma_f32_16x16x4_f32(false, a, false, b, (short)0, c, false, false);
}

// ------------------------------------------------------------------ zero ws
__global__ void k_zero(float* p, int n) {
  int i = blockIdx.x * blockDim.x + threadIdx.x;
  if (i < n) p[i] = 0.f;
}

// --------------------------------------------- hoisted input projections
// out[t][r][b] = sum_f W[r,f] * x[b,t,f]   (one 16x16 tile per wave)
__global__ void k_xproj(const float* __restrict__ W, const float* __restrict__ x,
                        float* __restrict__ out, int rows) {
  int lane = threadIdx.x;
  int m0 = blockIdx.x * 16;
  int j0 = blockIdx.y * 16;                 // column block over (t,b) pairs
  int col = j0 + (lane & 15);
  int t = col >> 6, b = col & 63;           // j = t*B + b
  const float* xcol = x + ((size_t)b * T_ + t) * F_;
  v8f acc = {};
  for (int k = 0; k < F_; k += 4) {
    v2f a = ld_a(W, F_, m0, k, lane);
    int kk = k + ((lane >> 4) << 1);
    v2f bf; bf.x = xcol[kk]; bf.y = xcol[kk + 1];
    acc = wmma4(a, bf, acc);
  }
  int hi = lane >> 4;
  int jc = j0 + (lane & 15); int tt = jc >> 6, bb = jc & 63;
  for (int v = 0; v < 8; ++v) {
    int r = m0 + v + 8 * hi;
    out[((size_t)tt * rows + r) * B_ + bb] = acc[v];
  }
}

// ------------------------------- per-step projections off h (+ scalar job)
// job 0: q_r=Wq_r@h+b   job 1: q_w   job 2: e=sigmoid(W_erase@h+b)
// job 3: cand=relu(W_ch@h + CP_t)    job 4: beta/gamma scalars
__global__ void k_proj(const float* __restrict__ hin,
                       const float* __restrict__ Wq_r, const float* __restrict__ bq_r,
                       const float* __restrict__ Wq_w, const float* __restrict__ bq_w,
                       const float* __restrict__ W_erase, const float* __restrict__ b_erase,
                       const float* __restrict__ W_ch, const float* __restrict__ cp_t,
                       const float* __restrict__ u_sr, const float* __restrict__ b_sr,
                       const float* __restrict__ u_lr, const float* __restrict__ b_lr,
                       const float* __restrict__ u_sw, const float* __restrict__ b_sw,
                       const float* __restrict__ u_lw, const float* __restrict__ b_lw,
                       float* __restrict__ q_r, float* __restrict__ q_w,
                       float* __restrict__ e_out, float* __restrict__ cand,
                       float* __restrict__ scal_out /* 4*B: bR,gR,bW,gW */) {
  int job = blockIdx.z;
  int lane = threadIdx.x;
  if (job == 4) {                               // tiny scalar gates
    if (blockIdx.x || blockIdx.y) return;
    const float* us[4] = {u_sr, u_lr, u_sw, u_lw};
    float bs[4] = {b_sr[0], b_lr[0], b_sw[0], b_lw[0]};
    for (int k = 0; k < 8; ++k) {
      int o = lane + 32 * k;                    // 0..255
      int sel = o >> 6, bb = o & 63;
      const float* u = us[sel];
      float s = 0.f;
      for (int i = 0; i < H_; ++i) s += u[i] * hin[i * B_ + bb];
      s += bs[sel];
      float outv = (sel & 1) ? (1.f / (1.f + expf(-s)))          // LRU gate: sigmoid
                             : (s > 20.f ? s : log1pf(expf(s))); // sharpen: softplus
      scal_out[sel * B_ + bb] = outv;
    }
    return;
  }
  int rows = (job < 2) ? M_ : C_;
  if (blockIdx.x * 16 >= rows) return;
  const float* A = (job == 0) ? Wq_r : (job == 1) ? Wq_w : (job == 2) ? W_erase : W_ch;
  int m0 = blockIdx.x * 16, n0 = blockIdx.y * 16;
  v8f acc = {};
  for (int k = 0; k < H_; k += 4) {
    v2f a = ld_a(A, H_, m0, k, lane);
    v2f b = ld_b(hin, B_, k, n0, lane);
    acc = wmma4(a, b, acc);
  }
  int hi = lane >> 4, c = n0 + (lane & 15);
  for (int v = 0; v < 8; ++v) {
    int i = m0 + v + 8 * hi;
    float x = acc[v];
    if (job == 0)      { x += bq_r[i];   q_r[i * B_ + c] = x; }
    else if (job == 1) { x += bq_w[i];   q_w[i * B_ + c] = x; }
    else if (job == 2) { x += b_erase[i]; e_out[i * B_ + c] = 1.f / (1.f + expf(-x)); }
    else               { x += cp_t[i * B_ + c]; cand[i * B_ + c] = fmaxf(x, 0.f); }
  }
}

// ---- addressing + softmax + read + memory update, one workgroup per batch
__global__ void k_mem(const float* __restrict__ q_r, const float* __restrict__ q_w,
                      const float* __restrict__ scal,
                      const float* __restrict__ e_in, const float* __restrict__ cand,
                      const float* __restrict__ addr,
                      float* __restrict__ Mc, float* __restrict__ ear,
                      float* __restrict__ eaw, float* __restrict__ reading) {
  __shared__ float qr_s[M_], qw_s[M_];
  __shared__ float ar_s[N_], aw_s[N_];
  __shared__ float red[256];
  __shared__ float sc[8];
  int tid = threadIdx.x, lane = tid & 31, warp = tid >> 5;
  int b = blockIdx.x;
  float beta_r = scal[0 * B_ + b], gamma_r = scal[1 * B_ + b];
  float beta_w = scal[2 * B_ + b], gamma_w = scal[3 * B_ + b];
  if (tid < 128) qr_s[tid] = q_r[tid * B_ + b];
  else           qw_s[tid - 128] = q_w[(tid - 128) * B_ + b];
  __syncthreads();
  if (warp < 2) {                                   // ||q||
    const float* qs = warp ? qw_s : qr_s;
    float s = 0.f;
    for (int m = lane; m < M_; m += 32) { float v = qs[m]; s += v * v; }
    for (int off = 16; off; off >>= 1) s += __shfl_xor(s, off);
    if (lane == 0) sc[warp] = sqrtf(s);
  }
  __syncthreads();
  float qn_r = sc[0], qn_w = sc[1];
  for (int n = warp; n < N_; n += 8) {              // sims for both heads
    const float* McRow = Mc + ((size_t)b * N_ + n) * C_;
    const float* adRow = addr + (size_t)n * A_;
    float dr = 0.f, dw = 0.f, nn = 0.f;
    for (int m = lane; m < M_; m += 32) {
      float v = (m < C_) ? McRow[m] : adRow[m - C_];
      dr += v * qr_s[m]; dw += v * qw_s[m]; nn += v * v;
    }
    for (int off = 16; off; off >>= 1) {
      dr += __shfl_xor(dr, off); dw += __shfl_xor(dw, off); nn += __shfl_xor(nn, off);
    }
    if (lane == 0) {
      float rn = sqrtf(nn);
      float s_r = beta_r * dr / (rn * qn_r + EPSF);
      float s_w = beta_w * dw / (rn * qn_w + EPSF);
      float er = ear[(size_t)b * N_ + n], ew = eaw[(size_t)b * N_ + n];
      ar_s[n] = s_r - gamma_r * er;
      aw_s[n] = s_w - gamma_w * ew;
      ear[(size_t)b * N_ + n] = 0.1f * er + 0.9f * s_r;
      eaw[(size_t)b * N_ + n] = 0.1f * ew + 0.9f * s_w;
    }
  }
  __syncthreads();
  // softmax: max
  float lr = -1e30f, lw = -1e30f;
  for (int n = tid; n < N_; n += 256) { lr = fmaxf(lr, ar_s[n]); lw = fmaxf(lw, aw_s[n]); }
  for (int off = 16; off; off >>= 1) {
    lr = fmaxf(lr, __shfl_xor(lr, off)); lw = fmaxf(lw, __shfl_xor(lw, off));
  }
  if (lane == 0) { red[warp] = lr; red[8 + warp] = lw; }
  __syncthreads();
  if (tid == 0) {
    float m1 = red[0], m2 = red[8];
    for (int i = 1; i < 8; ++i) { m1 = fmaxf(m1, red[i]); m2 = fmaxf(m2, red[8 + i]); }
    sc[2] = m1; sc[3] = m2;
  }
  __syncthreads();
  float mx_r = sc[2], mx_w = sc[3];
  float sr = 0.f, sw = 0.f;                          // exp + sum (store exp in place)
  for (int n = tid; n < N_; n += 256) {
    float er = expf(ar_s[n] - mx_r); ar_s[n] = er; sr += er;
    float ew = expf(aw_s[n] - mx_w); aw_s[n] = ew; sw += ew;
  }
  for (int off = 16; off; off >>= 1) { sr += __shfl_xor(sr, off); sw += __shfl_xor(sw, off); }
  if (lane == 0) { red[warp] = sr; red[8 + warp] = sw; }
  __syncthreads();
  if (tid == 0) {
    float s1 = 0.f, s2 = 0.f;
    for (int i = 0; i < 8; ++i) { s1 += red[i]; s2 += red[8 + i]; }
    sc[4] = s1; sc[5] = s2;
  }
  __syncthreads();
  float inv_r = 1.f / sc[4], inv_w = 1.f / sc[5];
  // fused: reading = M_full^T w_r ; Mc = Mc*(1-w_w e) + w_w cand  (single pass)
  int m = tid & 127, half = tid >> 7;
  float e_c = 0.f, cand_c = 0.f;
  if (m < C_) { e_c = e_in[m * B_ + b]; cand_c = cand[m * B_ + b]; }
  float racc = 0.f;
  for (int n = half; n < N_; n += 2) {
    float wr = ar_s[n] * inv_r;
    float v;
    if (m < C_) {
      size_t idx = ((size_t)b * N_ + n) * C_ + m;
      v = Mc[idx];
      float ww = aw_s[n] * inv_w;
      Mc[idx] = v * (1.f - ww * e_c) + ww * cand_c;
    } else {
      v = addr[(size_t)n * A_ + (m - C_)];
    }
    racc += v * wr;
  }
  __syncthreads();
  red[tid] = racc;
  __syncthreads();
  if (tid < 128) reading[tid * B_ + b] = red[tid] + red[tid + 128];
}

// ---------------------------- GRU step: one wave per 16x16 tile of h_new
__global__ void k_gru(const float* __restrict__ hin, float* __restrict__ hout,
                      const float* __restrict__ reading,
                      const float* __restrict__ W_hr, const float* __restrict__ W_mr,
                      const float* __restrict__ W_hz, const float* __restrict__ W_mz,
                      const float* __restrict__ W_hn, const float* __restrict__ W_mn,
                      const float* __restrict__ b_r, const float* __restrict__ b_z,
                      const float* __restrict__ b_n,
                      const float* __restrict__ xpr, const float* __restrict__ xpz,
                      const float* __restrict__ xpn) {
  int lane = threadIdx.x;
  int m0 = blockIdx.x * 16, n0 = blockIdx.y * 16;
  int hi = lane >> 4, c = n0 + (lane & 15);
  v8f acc_r, acc_z, acc_nm;                // init C from precomputed x-projections
  v8f acc_nh = {};
  for (int v = 0; v < 8; ++v) {
    int i = m0 + v + 8 * hi;
    acc_r[v]  = xpr[(size_t)i * B_ + c];
    acc_z[v]  = xpz[(size_t)i * B_ + c];
    acc_nm[v] = xpn[(size_t)i * B_ + c];
  }
  for (int k = 0; k < H_; k += 4) {        // shared B fragment from h
    v2f bh = ld_b(hin, B_, k, n0, lane);
    acc_r  = wmma4(ld_a(W_hr, H_, m0, k, lane), bh, acc_r);
    acc_z  = wmma4(ld_a(W_hz, H_, m0, k, lane), bh, acc_z);
    acc_nh = wmma4(ld_a(W_hn, H_, m0, k, lane), bh, acc_nh);
  }
  for (int k = 0; k < M_; k += 4) {        // shared B fragment from reading
    v2f bm = ld_b(reading, B_, k, n0, lane);
    acc_r  = wmma4(ld_a(W_mr, M_, m0, k, lane), bm, acc_r);
    acc_z  = wmma4(ld_a(W_mz, M_, m0, k, lane), bm, acc_z);
    acc_nm = wmma4(ld_a(W_mn, M_, m0, k, lane), bm, acc_nm);
  }
  for (int v = 0; v < 8; ++v) {
    int i = m0 + v + 8 * hi;
    float rg = 1.f / (1.f + expf(-(acc_r[v] + b_r[i])));
    float zg = 1.f / (1.f + expf(-(acc_z[v] + b_z[i])));
    float ng = tanhf(acc_nm[v] + b_n[i] + rg * acc_nh[v]);
    float ho = hin[(size_t)i * B_ + c];
    hout[(size_t)i * B_ + c] = (1.f - zg) * ng + zg * ho;
  }
}

// ------------------------------------------------------------- output proj
__global__ void k_out(const float* __restrict__ h, const float* __restrict__ W_out,
                      const float* __restrict__ b_out, float* __restrict__ out) {
  int tid = blockIdx.x * blockDim.x + threadIdx.x;
  if (tid >= B_ * O_) return;
  int b = tid / O_, o = tid % O_;
  float s = b_out[o];
  for (int i = 0; i < H_; ++i) s += h[i * B_ + b] * W_out[o * H_ + i];
  out[(size_t)b * O_ + o] = s;
}

extern "C" void kernel_launch(void* const* d_in, const int* in_sizes, int n_in,
                              void* d_out, int out_size, void* d_ws, size_t ws_size,
                              hipStream_t stream) {
  (void)in_sizes; (void)n_in; (void)out_size; (void)ws_size;
  const float* x         = (const float*)d_in[0];
  const float* addresses = (const float*)d_in[1];
  const float* W_output  = (const float*)d_in[2];
  const float* b_output  = (const float*)d_in[3];
  const float* W_xr = (const float*)d_in[4];
  const float* W_hr = (const float*)d_in[5];
  const float* W_mr = (const float*)d_in[6];
  const float* b_r  = (const float*)d_in[7];
  const float* W_xz = (const float*)d_in[8];
  const float* W_hz = (const float*)d_in[9];
  const float* W_mz = (const float*)d_in[10];
  const float* b_z  = (const float*)d_in[11];
  const float* W_xn = (const float*)d_in[12];
  const float* W_hn = (const float*)d_in[13];
  const float* W_mn = (const float*)d_in[14];
  const float* b_n  = (const float*)d_in[15];
  const float* W_query_r = (const float*)d_in[16];
  const float* b_query_r = (const float*)d_in[17];
  const float* u_sr = (const float*)d_in[18];
  const float* b_sr = (const float*)d_in[19];
  const float* u_lr = (const float*)d_in[20];
  const float* b_lr = (const float*)d_in[21];
  const float* W_query_w = (const float*)d_in[22];
  const float* b_query_w = (const float*)d_in[23];
  const float* u_sw = (const float*)d_in[24];
  const float* b_sw = (const float*)d_in[25];
  const float* u_lw = (const float*)d_in[26];
  const float* b_lw = (const float*)d_in[27];
  const float* W_erase = (const float*)d_in[28];
  const float* b_erase = (const float*)d_in[29];
  const float* W_ch = (const float*)d_in[30];
  const float* W_ci = (const float*)d_in[31];

  float* ws = (float*)d_ws;
  size_t off = 0;
  float* h0  = ws + off; off += H_ * B_;
  float* Mc  = ws + off; off += (size_t)B_ * N_ * C_;
  float* ear = ws + off; off += (size_t)B_ * N_;
  float* eaw = ws + off; off += (size_t)B_ * N_;
  size_t zero_len = off;                       // h0,Mc,ear,eaw contiguous
  float* h1   = ws + off; off += H_ * B_;
  float* reading = ws + off; off += M_ * B_;
  float* q_r  = ws + off; off += M_ * B_;
  float* q_w  = ws + off; off += M_ * B_;
  float* e_v  = ws + off; off += C_ * B_;
  float* cand = ws + off; off += C_ * B_;
  float* scal = ws + off; off += 4 * B_;
  float* XPr  = ws + off; off += (size_t)T_ * H_ * B_;
  float* XPz  = ws + off; off += (size_t)T_ * H_ * B_;
  float* XPn  = ws + off; off += (size_t)T_ * H_ * B_;
  float* CP   = ws + off; off += (size_t)T_ * C_ * B_;

  k_zero<<<(int)((zero_len + 255) / 256), 256, 0, stream>>>(h0, (int)zero_len);
  // hoisted input projections (no recurrence)
  k_xproj<<<dim3(H_ / 16, (T_ * B_) / 16), 32, 0, stream>>>(W_xr, x, XPr, H_);
  k_xproj<<<dim3(H_ / 16, (T_ * B_) / 16), 32, 0, stream>>>(W_xz, x, XPz, H_);
  k_xproj<<<dim3(H_ / 16, (T_ * B_) / 16), 32, 0, stream>>>(W_xn, x, XPn, H_);
  k_xproj<<<dim3(C_ / 16, (T_ * B_) / 16), 32, 0, stream>>>(W_ci, x, CP, C_);

  float* hin = h0; float* hout = h1;
  for (int t = 0; t < T_; ++t) {
    k_proj<<<dim3(8, 4, 5), 32, 0, stream>>>(hin,
        W_query_r, b_query_r, W_query_w, b_query_w,
        W_erase, b_erase, W_ch, CP + (size_t)t * C_ * B_,
        u_sr, b_sr, u_lr, b_lr, u_sw, b_sw, u_lw, b_lw,
        q_r, q_w, e_v, cand, scal);
    k_mem<<<B_, 256, 0, stream>>>(q_r, q_w, scal, e_v, cand, addresses,
                                  Mc, ear, eaw, reading);
    k_gru<<<dim3(H_ / 16, B_ / 16), 32, 0, stream>>>(hin, hout, reading,
        W_hr, W_mr, W_hz, W_mz, W_hn, W_mn, b_r, b_z, b_n,
        XPr + (size_t)t * H_ * B_, XPz + (size_t)t * H_ * B_,
        XPn + (size_t)t * H_ * B_);
    float* tmp = hin; hin = hout; hout = tmp;
  }
  k_out<<<(B_ * O_ + 255) / 256, 256, 0, stream>>>(hin, W_output, b_output,
                                                   (float*)d_out);
}